// TripletSelection_72911364817292
// MI455X (gfx1250) — compile-verified
//
#include <hip/hip_runtime.h>
#include <hip/hip_bf16.h>
#include <stdint.h>

// ---------------------------------------------------------------------------
// Triplet Attention fused for MI455X (gfx1250, wave32).
// B=8, C=256, H=W=64, fp32.  Gates are low-rank => never materialize branches.
// ---------------------------------------------------------------------------

#define Bn 8
#define Cn 256
#define Hn 64
#define Wn 64
#define PLANE (Hn * Wn)          // 4096

typedef float v2f __attribute__((ext_vector_type(2)));
typedef float v8f __attribute__((ext_vector_type(8)));

// ---------------- ZPool kernels (max+mean over one axis) -------------------

// reduce over C -> zp3 [B,2,H,W]
__global__ void zpool_c_kernel(const float* __restrict__ x, float* __restrict__ zp3) {
    int flat = blockIdx.x * 256 + threadIdx.x;          // B*H*W = 32768
    int w = flat & 63, h = (flat >> 6) & 63, b = flat >> 12;
    const float* xp = x + (size_t)b * Cn * PLANE + h * Wn + w;
    float mx = -3.4e38f, sm = 0.f;
    for (int c = 0; c < Cn; ++c) {
        float v = xp[(size_t)c * PLANE];
        mx = fmaxf(mx, v); sm += v;
    }
    size_t base = (size_t)b * 2 * PLANE + h * Wn + w;
    zp3[base] = mx;
    zp3[base + PLANE] = sm * (1.0f / Cn);
}

// reduce over H -> zp1 [B,2,C,W]
__global__ void zpool_h_kernel(const float* __restrict__ x, float* __restrict__ zp1) {
    int flat = blockIdx.x * 256 + threadIdx.x;          // B*C*W = 131072
    int w = flat & 63, c = (flat >> 6) & 255, b = flat >> 14;
    const float* xp = x + ((size_t)(b * Cn + c)) * PLANE + w;
    float mx = -3.4e38f, sm = 0.f;
    for (int h = 0; h < Hn; ++h) {
        float v = xp[h * Wn];
        mx = fmaxf(mx, v); sm += v;
    }
    size_t base = ((size_t)(b * 2) * Cn + c) * Wn + w;  // ch stride = C*W
    zp1[base] = mx;
    zp1[base + Cn * Wn] = sm * (1.0f / Hn);
}

// reduce over W -> zp2 [B,2,H,C]
__global__ void zpool_w_kernel(const float* __restrict__ x, float* __restrict__ zp2) {
    int flat = blockIdx.x * 256 + threadIdx.x;          // B*H*C = 131072, c fastest
    int c = flat & 255, h = (flat >> 8) & 63, b = flat >> 14;
    const float4* xp = (const float4*)(x + ((size_t)(b * Cn + c)) * PLANE + h * Wn);
    float mx = -3.4e38f, sm = 0.f;
    for (int i = 0; i < 16; ++i) {
        float4 v = xp[i];
        mx = fmaxf(mx, fmaxf(fmaxf(v.x, v.y), fmaxf(v.z, v.w)));
        sm += v.x + v.y + v.z + v.w;
    }
    size_t base = ((size_t)(b * 2) * Hn + h) * Cn + c;  // ch stride = H*C
    zp2[base] = mx;
    zp2[base + Hn * Cn] = sm * (1.0f / Wn);
}

// ------------- 7x7 conv (2ch->1) + BN + sigmoid via f32 WMMA ---------------
// One wave computes 16 outputs along Q for a fixed (nb, p) row.
// Input tile (2 ch x 7 rows x 22 cols, zero-padded halo) is staged in LDS, so
// the fully-unrolled K loop is: 2 ds_loads + cndmask per operand -> v_wmma.
// im2col: K = 2*7*7 = 98, padded to 100, tiled K=4 per V_WMMA_F32_16X16X4_F32.
// A[m=col][k] = tile patch; B[k][n] = filter tap broadcast over all 16 cols.
// D column 0 (lanes 0/16, 8 VGPRs each) = the 16 conv outputs.

__device__ __forceinline__ constexpr int koff(int k) {
    // k -> (ci, r, s); LDS tile offset for output col 0 (folds at compile time)
    int ci = k / 49;
    int rem = k - ci * 49;
    int r = rem / 7;
    int s = rem - r * 7;
    return (ci * 7 + r) * 22 + s;
}

__global__ void gate_conv_wmma_kernel(const float* __restrict__ zp,    // [NB,2,P,Q]
                                      const float* __restrict__ wconv, // 98 floats
                                      const float* __restrict__ gbn,
                                      const float* __restrict__ bbn,
                                      const float* __restrict__ mbn,
                                      const float* __restrict__ vbn,
                                      float* __restrict__ gate,        // [NB,P,Q]
                                      int NB, int P, int Q) {
    __shared__ float wsm[100];
    __shared__ float tile[308];               // [2][7][22] zero-padded halo
    int lane = threadIdx.x;                   // blockDim.x == 32 (one wave)
    for (int i = lane; i < 100; i += 32) wsm[i] = (i < 98) ? wconv[i] : 0.0f;

    int qt = Q >> 4;
    int t = blockIdx.x;
    int q0 = (t % qt) << 4;
    int p  = (t / qt) % P;
    int nb = t / (qt * P);
    const float* zb = zp + (size_t)nb * 2 * P * Q;

    // stage halo tile (all bounds checks live here, once)
    for (int i = lane; i < 308; i += 32) {
        int ci = i / 154, rem = i - ci * 154;
        int r = rem / 22, cc = rem - r * 22;
        int pp = p + r - 3, qq = q0 + cc - 3;
        float v = 0.0f;
        if ((unsigned)pp < (unsigned)P && (unsigned)qq < (unsigned)Q)
            v = zb[((size_t)ci * P + pp) * Q + qq];
        tile[i] = v;
    }
    __syncthreads();

    int col  = lane & 15;                     // A-matrix row m (output q0+col)
    int half = lane >> 4;                     // K sub-pair selector

    v8f acc = {};
#pragma unroll
    for (int k0 = 0; k0 < 100; k0 += 4) {
        v2f a, bv;
#pragma unroll
        for (int j = 0; j < 2; ++j) {
            int klo = k0 + j;                 // constant after unroll
            int khi = klo + 2;                // constant after unroll
            int klo_c = (klo > 97) ? 97 : klo;  // clamp (tap weight is 0 there)
            int khi_c = (khi > 97) ? 97 : khi;
            int off = col + (half ? koff(khi_c) : koff(klo_c));
            a[j]  = tile[off];
            bv[j] = wsm[klo + 2 * half];      // zero-padded for k >= 98
        }
        // D = A(16x4) * B(4x16) + C ; fp32 matrix pipe
        acc = __builtin_amdgcn_wmma_f32_16x16x4_f32(
                  false, a, false, bv, (short)0, acc, false, false);
    }

    if (col == 0) {                           // lanes 0 and 16 hold N=0 column
        float scale = gbn[0] * __frsqrt_rn(vbn[0] + 1e-5f);
        float mm = mbn[0], bb = bbn[0];
        float* gout = gate + ((size_t)nb * P + p) * Q;
#pragma unroll
        for (int g = 0; g < 8; ++g) {         // M = g + 8*half
            float y = (acc[g] - mm) * scale + bb;
            gout[q0 + g + 8 * half] = 1.0f / (1.0f + __expf(-y));
        }
    }
}

// ------------- pooled[b,c] = mean_{h,w} x*(g1+g2+g3)/3 ---------------------
__global__ void pooled_kernel(const float* __restrict__ x,
                              const float* __restrict__ g1,   // [B,C,W]
                              const float* __restrict__ g2,   // [B,H,C]
                              const float* __restrict__ g3,   // [B,H,W]
                              float* __restrict__ pooled) {   // [B,C]
    __shared__ float red[256];
    int tid = threadIdx.x, blk = blockIdx.x;   // blk = b*C + c
    int b = blk >> 8, c = blk & 255;
    const float* xp  = x  + (size_t)blk * PLANE;
    const float* g1p = g1 + (size_t)blk * Wn;
    const float* g3p = g3 + (size_t)b * PLANE;
    float s = 0.f;
    for (int i = tid; i < PLANE; i += 256) {
        int h = i >> 6, w = i & 63;
        s += xp[i] * (g1p[w] + g2[((size_t)(b * Hn + h)) * Cn + c] + g3p[i]);
    }
    red[tid] = s; __syncthreads();
    for (int st = 128; st > 0; st >>= 1) {
        if (tid < st) red[tid] += red[tid + st];
        __syncthreads();
    }
    if (tid == 0) pooled[blk] = red[0] * (1.0f / (3.0f * PLANE));
}

// ------------- conv1d(k=7, 3 filters) over C + softmax --------------------
__global__ void weights_kernel(const float* __restrict__ pooled, // [B,C]
                               const float* __restrict__ w1d,    // [3,1,7]
                               float* __restrict__ wt) {         // [B,3,C]
    int c = threadIdx.x, b = blockIdx.x;
    float l[3];
#pragma unroll
    for (int o = 0; o < 3; ++o) {
        float s = 0.f;
#pragma unroll
        for (int k = 0; k < 7; ++k) {
            int cc = c + k - 3;
            if ((unsigned)cc < (unsigned)Cn) s += pooled[b * Cn + cc] * w1d[o * 7 + k];
        }
        l[o] = s;
    }
    float m = fmaxf(l[0], fmaxf(l[1], l[2]));
    float e0 = __expf(l[0] - m), e1 = __expf(l[1] - m), e2 = __expf(l[2] - m);
    float inv = 1.0f / (e0 + e1 + e2);
    wt[(b * 3 + 0) * Cn + c] = e0 * inv;
    wt[(b * 3 + 1) * Cn + c] = e1 * inv;
    wt[(b * 3 + 2) * Cn + c] = e2 * inv;
}

// ------------- final combine: out = x*(w1*g1 + w2*g2 + w3*g3) --------------
// Stages the 16KB x-plane into LDS with async b128 copies (ASYNCcnt path).
__global__ void combine_kernel(const float* __restrict__ x,
                               const float* __restrict__ g1,
                               const float* __restrict__ g2,
                               const float* __restrict__ g3,
                               const float* __restrict__ wt,
                               float* __restrict__ out) {
    __shared__ float xs[PLANE];
    __shared__ float g1s[Wn];
    __shared__ float g2s[Hn];
    int tid = threadIdx.x, blk = blockIdx.x;   // blk = b*C + c
    int b = blk >> 8, c = blk & 255;
    const float* xp = x + (size_t)blk * PLANE;

    // async-stage 4096 floats: 256 threads x 4 b128 ops
#pragma unroll
    for (int i = 0; i < 4; ++i) {
        int e = tid * 4 + i * 1024;
        unsigned lds_off =
            (unsigned)(unsigned long long)(__attribute__((address_space(3))) float*)&xs[e];
        unsigned long long ga = (unsigned long long)(uintptr_t)(xp + e);
        asm volatile("global_load_async_to_lds_b128 %0, %1, off"
                     :: "v"(lds_off), "v"(ga) : "memory");
    }
    if (tid < 64) {
        g1s[tid] = g1[(size_t)blk * Wn + tid];
        g2s[tid] = g2[((size_t)(b * Hn + tid)) * Cn + c];
    }
    float w1 = wt[(b * 3 + 0) * Cn + c];
    float w2 = wt[(b * 3 + 1) * Cn + c];
    float w3 = wt[(b * 3 + 2) * Cn + c];

    asm volatile("s_wait_asynccnt 0" ::: "memory");
    __syncthreads();

    const float* g3b = g3 + (size_t)b * PLANE;
    float* op = out + (size_t)blk * PLANE;
#pragma unroll
    for (int i = 0; i < 16; ++i) {
        int e = tid + i * 256;
        int h = e >> 6, w = e & 63;
        float coeff = w1 * g1s[w] + w2 * g2s[h] + w3 * g3b[e];
        op[e] = xs[e] * coeff;
    }
}

// ---------------------------------------------------------------------------

extern "C" void kernel_launch(void* const* d_in, const int* in_sizes, int n_in,
                              void* d_out, int out_size, void* d_ws, size_t ws_size,
                              hipStream_t stream) {
    const float* x    = (const float*)d_in[0];
    const float* w_cw = (const float*)d_in[1];   // branch 1 (g1)
    const float* g_cw = (const float*)d_in[2];
    const float* b_cw = (const float*)d_in[3];
    const float* m_cw = (const float*)d_in[4];
    const float* v_cw = (const float*)d_in[5];
    const float* w_hc = (const float*)d_in[6];   // branch 2 (g2)
    const float* g_hc = (const float*)d_in[7];
    const float* b_hc = (const float*)d_in[8];
    const float* m_hc = (const float*)d_in[9];
    const float* v_hc = (const float*)d_in[10];
    const float* w_hw = (const float*)d_in[11];  // branch 3 (g3)
    const float* g_hw = (const float*)d_in[12];
    const float* b_hw = (const float*)d_in[13];
    const float* m_hw = (const float*)d_in[14];
    const float* v_hw = (const float*)d_in[15];
    const float* w1d  = (const float*)d_in[16];
    float* out = (float*)d_out;

    // workspace layout (floats)
    float* ws     = (float*)d_ws;
    float* zp1    = ws;                    // [B,2,C,W]  262144
    float* zp2    = zp1 + 262144;          // [B,2,H,C]  262144
    float* zp3    = zp2 + 262144;          // [B,2,H,W]   65536
    float* g1     = zp3 + 65536;           // [B,C,W]    131072
    float* g2     = g1  + 131072;          // [B,H,C]    131072
    float* g3     = g2  + 131072;          // [B,H,W]     32768
    float* pooled = g3  + 32768;           // [B,C]        2048
    float* wt     = pooled + 2048;         // [B,3,C]      6144

    zpool_c_kernel<<<128, 256, 0, stream>>>(x, zp3);
    zpool_h_kernel<<<512, 256, 0, stream>>>(x, zp1);
    zpool_w_kernel<<<512, 256, 0, stream>>>(x, zp2);

    // gate convs (WMMA f32): grid = NB*P*(Q/16) waves
    gate_conv_wmma_kernel<<<Bn * Cn * (Wn / 16), 32, 0, stream>>>(
        zp1, w_cw, g_cw, b_cw, m_cw, v_cw, g1, Bn, Cn, Wn);
    gate_conv_wmma_kernel<<<Bn * Hn * (Cn / 16), 32, 0, stream>>>(
        zp2, w_hc, g_hc, b_hc, m_hc, v_hc, g2, Bn, Hn, Cn);
    gate_conv_wmma_kernel<<<Bn * Hn * (Wn / 16), 32, 0, stream>>>(
        zp3, w_hw, g_hw, b_hw, m_hw, v_hw, g3, Bn, Hn, Wn);

    pooled_kernel<<<Bn * Cn, 256, 0, stream>>>(x, g1, g2, g3, pooled);
    weights_kernel<<<Bn, 256, 0, stream>>>(pooled, w1d, wt);
    combine_kernel<<<Bn * Cn, 256, 0, stream>>>(x, g1, g2, g3, wt, out);
}